// ZBML3_85366769975906
// MI455X (gfx1250) — compile-verified
//
#include <hip/hip_runtime.h>
#include <math.h>
#include <stdint.h>

typedef float v2f __attribute__((ext_vector_type(2)));
typedef float v8f __attribute__((ext_vector_type(8)));

#define T_ 16
#define Z_ 33
#define B_ 128
#define FV_ 784
#define M_ 8
#define NSTEP_ 4
#define K_ 32
#define SEP_ 50   // FV//T + 1
#define KP_ 52    // K padded to multiple of 4
#define ZP_ 48    // Z padded to multiple of 16 (also lp_seg row padding)

// ---------------- Threefry-2x32 (JAX-compatible) ----------------
__device__ __forceinline__ uint32_t rotl32(uint32_t v, int r) {
  return (v << r) | (v >> (32 - r));
}

__device__ __forceinline__ void threefry2x32(uint32_t k0, uint32_t k1,
                                             uint32_t x0, uint32_t x1,
                                             uint32_t& o0, uint32_t& o1) {
  uint32_t ks2 = k0 ^ k1 ^ 0x1BD11BDAu;
  x0 += k0; x1 += k1;
#define TFR(r) { x0 += x1; x1 = rotl32(x1, r); x1 ^= x0; }
  TFR(13) TFR(15) TFR(26) TFR(6)   x0 += k1;  x1 += ks2 + 1u;
  TFR(17) TFR(29) TFR(16) TFR(24)  x0 += ks2; x1 += k0  + 2u;
  TFR(13) TFR(15) TFR(26) TFR(6)   x0 += k0;  x1 += k1  + 3u;
  TFR(17) TFR(29) TFR(16) TFR(24)  x0 += k1;  x1 += ks2 + 4u;
  TFR(13) TFR(15) TFR(26) TFR(6)   x0 += ks2; x1 += k0  + 5u;
#undef TFR
  o0 = x0; o1 = x1;
}

// ---------------- Kernel 1: lp_seg[t][z][b] via f32 WMMA ----------------
// lp_seg[t,z,b] = c2*(sum x^2 + sum loc^2) + (1/s^2)*<loc, x> + cst
// Tiles staged zero-padded in LDS once; WMMA loop is branch-free with two
// independent accumulators; epilogue stores unconditionally into the
// z-padded (t*48+z) workspace layout.
__global__ void __launch_bounds__(256)
k_lp_seg(const float* __restrict__ img, const float* __restrict__ whz,
         const float* __restrict__ bias, const float* __restrict__ bso,
         float* __restrict__ lp_seg) {
  const int t = blockIdx.x;
  const int tid = threadIdx.x;
  const int segbase = t * SEP_;
  const int seglen = min(FV_, (t + 1) * SEP_) - segbase;   // 50, last slot 34
  const float bv = bias[0];
  const float s = expf(bso[0]);
  const float inv_s2 = 1.0f / (s * s);
  const float c2 = -0.5f * inv_s2;
  const float cst = -(float)seglen * (logf(s) + 0.91893853320467274f); // 0.5*log(2pi)

  __shared__ float sA[ZP_ * KP_];     // loc tile, zero padded   (48x52)
  __shared__ float sBt[B_ * KP_];     // x tile, b-major, padded (128x52)
  __shared__ float s_sumx2[B_];
  __shared__ float s_sumloc2[ZP_];

  // ---- stage tiles into LDS (one-time, coherent guards) ----
  for (int idx = tid; idx < ZP_ * KP_; idx += 256) {
    const int z = idx / KP_, j = idx - z * KP_;
    float v = 0.f;
    if (z < Z_ && j < seglen) v = whz[(t * Z_ + z) * FV_ + segbase + j] + bv;
    sA[idx] = v;
  }
  for (int idx = tid; idx < B_ * KP_; idx += 256) {
    const int b = idx / KP_, j = idx - b * KP_;
    float v = 0.f;
    if (j < seglen) v = img[b * FV_ + segbase + j];
    sBt[idx] = v;
  }
  __syncthreads();

  // ---- rank-1 correction sums straight from the padded LDS tiles ----
  if (tid < B_) {
    float acc = 0.f;
    const float* r = &sBt[tid * KP_];
    for (int j = 0; j < KP_; ++j) acc += r[j] * r[j];
    s_sumx2[tid] = acc;
  } else if (tid < B_ + ZP_) {
    const int z = tid - B_;
    float acc = 0.f;
    const float* r = &sA[z * KP_];
    for (int j = 0; j < KP_; ++j) acc += r[j] * r[j];
    s_sumloc2[z] = acc;
  }
  __syncthreads();

  const int lane = tid & 31;
  const int wave = tid >> 5;
  const int half = lane >> 4;      // which 16-lane half
  const int l15  = lane & 15;

  // 3 z-tiles x 8 b-tiles = 24 output tiles, 8 waves -> 3 tiles each
  for (int tile = wave; tile < 24; tile += 8) {
    const int zt = tile >> 3;                 // 0..2
    const int bt = tile & 7;                  // 0..7
    const int zrow = zt * 16 + l15;           // A row (M), padded rows are zero
    const int bcol = bt * 16 + l15;           // B col (N)
    const float* pa = &sA[zrow * KP_];
    const float* pb = &sBt[bcol * KP_];

    v8f acc0 = {}, acc1 = {};                 // two chains -> WMMA ILP
#pragma unroll
    for (int ks = 0; ks < 12; ks += 2) {      // K-steps 0..11 split even/odd
      const int j0 = ks * 4 + half * 2;
      const v2f a0 = *(const v2f*)(pa + j0);
      const v2f b0 = *(const v2f*)(pb + j0);
      acc0 = __builtin_amdgcn_wmma_f32_16x16x4_f32(false, a0, false, b0,
                                                   (short)0, acc0, false, false);
      const v2f a1 = *(const v2f*)(pa + j0 + 4);
      const v2f b1 = *(const v2f*)(pb + j0 + 4);
      acc1 = __builtin_amdgcn_wmma_f32_16x16x4_f32(false, a1, false, b1,
                                                   (short)0, acc1, false, false);
    }
    {                                          // K-step 12 (j = 48..51)
      const int j0 = 48 + half * 2;
      const v2f a = *(const v2f*)(pa + j0);
      const v2f b = *(const v2f*)(pb + j0);
      acc0 = __builtin_amdgcn_wmma_f32_16x16x4_f32(false, a, false, b,
                                                   (short)0, acc0, false, false);
    }

    const float sx = s_sumx2[bcol];            // loop-invariant, hoisted
    const int zb = zt * 16 + half * 8;         // D layout: lanes>=16 hold M=r+8
#pragma unroll
    for (int r = 0; r < 8; ++r) {
      const int z = zb + r;
      const float lp = c2 * (sx + s_sumloc2[z])
                     + inv_s2 * (acc0[r] + acc1[r]) + cst;
      lp_seg[(t * ZP_ + z) * B_ + bcol] = lp;  // padded rows never read
    }
  }
}

// ---------------- Kernel 2: lp_prior_log = log_softmax(prior, -1) ----------------
__global__ void k_log_softmax(const float* __restrict__ prior, float* __restrict__ out) {
  const int rid = blockIdx.x * blockDim.x + threadIdx.x;   // 0..T_*Z_-1
  if (rid >= T_ * Z_) return;
  const float* r = prior + rid * Z_;
  float mx = r[0];
  for (int j = 1; j < Z_; ++j) mx = fmaxf(mx, r[j]);
  float sum = 0.f;
  for (int j = 0; j < Z_; ++j) sum += expf(r[j] - mx);
  const float lse = mx + logf(sum);
  float* o = out + rid * Z_;
  for (int j = 0; j < Z_; ++j) o[j] = r[j] - lse;
}

// ---------------- Kernel 3: 64-step Gibbs scan ----------------
// One block per batch b; wave m (of 8) = chain (b,m); lane k = candidate k (K=32=wave32).
// Slot states distributed: lane j<16 holds x[j]; read via __shfl.
__global__ void __launch_bounds__(256)
k_scan(const float* __restrict__ lp_seg, const float* __restrict__ lp_prior,
       float* __restrict__ lp_m) {
  const int b = blockIdx.x;
  const int tid = threadIdx.x;
  const int m = tid >> 5;
  const int lane = tid & 31;

  __shared__ float s_lp[T_ * Z_];            // per-b lp_seg slice, 528 floats
  for (int i = tid; i < T_ * Z_; i += 256) {
    const int t = i / Z_, z = i - t * Z_;
    s_lp[i] = lp_seg[(t * ZP_ + z) * B_ + b];
  }
  __syncthreads();

  int xslot = 0;                              // lane j<16: state of slot j (init 0)
  float base_ext = 0.f, base_int = 0.f;
  for (int t = 0; t < T_; ++t) {
    base_ext += s_lp[t * Z_];                 // lp_seg[t,0,b]
    base_int += lp_prior[t * Z_ * Z_];        // lp_prior[t,0,0]
  }

  float lp_sel = 0.f;
  const uint32_t f = (uint32_t)((b * M_ + m) * K_ + lane); // flat index in (B,M,K)
  const uint32_t HALFN = (uint32_t)(B_ * M_ * K_ / 2);     // 16384

  for (int i = 0; i < NSTEP_ * T_; ++i) {
    const int t = i & (T_ - 1);
    const int xt    = __shfl(xslot, t, 32);
    const int xprev = (t > 0)      ? __shfl(xslot, t - 1, 32) : 0;
    const int xnext = (t < T_ - 1) ? __shfl(xslot, t + 1, 32) : 0;

    const float ext_minus = base_ext - s_lp[t * Z_ + xt];
    const float* Prow_t = lp_prior + (t * Z_ + xprev) * Z_; // row: slot t, parent xprev
    const float term_t = Prow_t[xt];
    float term_n = 0.f, cand_n = 0.f;
    if (t < T_ - 1) {
      const float* Pn = lp_prior + ((t + 1) * Z_) * Z_;
      term_n = Pn[xt * Z_ + xnext];           // old: parent xt -> child xnext
      cand_n = Pn[lane * Z_ + xnext];         // cand: parent k  -> child xnext
    }
    const float int_minus = base_int - term_t - term_n;

    const float cand_ext = s_lp[t * Z_ + lane];
    const float cand_t = Prow_t[lane];
    const float lp_tot = ext_minus + cand_ext + int_minus + cand_t + cand_n;

    // key_i = fold_in(key(42), i) = threefry((0,42), (0,i))
    uint32_t kk0, kk1;
    threefry2x32(0u, 42u, 0u, (uint32_t)i, kk0, kk1);
    // JAX random_bits split layout over n=32768 counters
    uint32_t o0, o1, bits;
    if (f < HALFN) { threefry2x32(kk0, kk1, f, f + HALFN, o0, o1); bits = o0; }
    else           { threefry2x32(kk0, kk1, f - HALFN, f, o0, o1); bits = o1; }
    const float TINY = 1.1754943508222875e-38f;
    const float fl = __uint_as_float((bits >> 9) | 0x3f800000u) - 1.0f;
    const float u = fmaxf(TINY, fl * (1.0f - TINY) + TINY);
    const float g = -logf(-logf(u));          // gumbel

    // wave-wide argmax of lp_tot + g, ties -> lowest lane (matches jnp.argmax)
    float best_s = lp_tot + g, best_lp = lp_tot;
    int best_k = lane;
#pragma unroll
    for (int off = 16; off > 0; off >>= 1) {
      const float os = __shfl_xor(best_s, off, 32);
      const float ol = __shfl_xor(best_lp, off, 32);
      const int   ok = __shfl_xor(best_k, off, 32);
      if (os > best_s || (os == best_s && ok < best_k)) {
        best_s = os; best_lp = ol; best_k = ok;
      }
    }
    base_ext = ext_minus + __shfl(cand_ext, best_k, 32);
    base_int = int_minus + __shfl(cand_t + cand_n, best_k, 32);
    if (lane == t) xslot = best_k;
    lp_sel = best_lp;
  }
  if (lane == 0) lp_m[b * M_ + m] = lp_sel;
}

// ---------------- Kernel 4: -mean_b logsumexp_m ----------------
__global__ void __launch_bounds__(128)
k_reduce(const float* __restrict__ lp_m, float* __restrict__ out) {
  const int b = threadIdx.x;
  __shared__ float s_lse[B_];
  const float* r = lp_m + b * M_;
  float mx = r[0];
  for (int j = 1; j < M_; ++j) mx = fmaxf(mx, r[j]);
  float sum = 0.f;
  for (int j = 0; j < M_; ++j) sum += expf(r[j] - mx);
  s_lse[b] = mx + logf(sum);
  __syncthreads();
  for (int off = 64; off > 0; off >>= 1) {
    if (b < off) s_lse[b] += s_lse[b + off];
    __syncthreads();
  }
  if (b == 0) out[0] = -s_lse[0] / (float)B_;
}

// ---------------- Launch ----------------
extern "C" void kernel_launch(void* const* d_in, const int* in_sizes, int n_in,
                              void* d_out, int out_size, void* d_ws, size_t ws_size,
                              hipStream_t stream) {
  const float* img   = (const float*)d_in[0];   // (128,28,28)
  const float* whz   = (const float*)d_in[1];   // (16,33,784)
  const float* bias  = (const float*)d_in[2];   // (1,)
  const float* bso   = (const float*)d_in[3];   // scalar
  const float* prior = (const float*)d_in[4];   // (16,33,33)
  (void)in_sizes; (void)n_in; (void)out_size; (void)ws_size;

  float* ws = (float*)d_ws;
  float* lp_seg   = ws;                         // 16*48*128 = 98304 floats (z-padded)
  float* lp_prior = lp_seg + T_ * ZP_ * B_;     // 16*33*33  = 17424 floats
  float* lp_m     = lp_prior + T_ * Z_ * Z_;    // 128*8     = 1024 floats

  k_lp_seg     <<<T_, 256, 0, stream>>>(img, whz, bias, bso, lp_seg);
  k_log_softmax<<<(T_ * Z_ + 255) / 256, 256, 0, stream>>>(prior, lp_prior);
  k_scan       <<<B_, 256, 0, stream>>>(lp_seg, lp_prior, lp_m);
  k_reduce     <<<1, B_, 0, stream>>>(lp_m, (float*)d_out);
}